// blockNL_3633542332810
// MI455X (gfx1250) — compile-verified
//
#include <hip/hip_runtime.h>

typedef __attribute__((ext_vector_type(2))) float v2f;
typedef __attribute__((ext_vector_type(8))) float v8f;

#define HH 256
#define WW 256
#define NB 4
#define NC 4
#define FSZ 11
#define PADR 5
#define QT 4               // 4x4 query tile per wave
#define KRR 14             // QT + FSZ - 1 (key region side)
#define NKEY 196           // KRR*KRR
#define NKP 208            // padded to multiple of 16

__global__ __launch_bounds__(32)
void blocknl_wmma_kernel(const float* __restrict__ x,
                         const float* __restrict__ Wt,
                         const float* __restrict__ Wp,
                         const float* __restrict__ Wg,
                         const float* __restrict__ Ww,
                         float* __restrict__ out)
{
    __shared__ float th[NC][NKP];   // theta rows over key region
    __shared__ float ph[NC][NKP];   // phi rows
    __shared__ float gg[NC][NKP];   // g rows
    __shared__ float sc[16][NKP];   // raw scores -> exp weights (unnormalized)
    __shared__ float ov[NC][16];    // attended output (pre-Ww)

    const int lane = threadIdx.x;   // 0..31, wave32
    const int lm   = lane & 15;
    const int hi   = lane >> 4;
    const int k0   = hi * 2;        // K-slice this lane supplies (A & B share it)

    int t = blockIdx.x;
    const int bb = t >> 12;         // / (64*64)
    t &= 4095;
    const int h0 = (t >> 6) * QT;
    const int w0 = (t & 63) * QT;

    const v8f cz = {};

    // ---- Stage A: [Wt;Wp;Wg] @ x over the 14x14 key region (zero-padded) ----
    float a0 = 0.f, a1 = 0.f;
    if (lm < 4)       { a0 = Wt[lm*4 + k0];      a1 = Wt[lm*4 + k0 + 1]; }
    else if (lm < 8)  { a0 = Wp[(lm-4)*4 + k0];  a1 = Wp[(lm-4)*4 + k0 + 1]; }
    else if (lm < 12) { a0 = Wg[(lm-8)*4 + k0];  a1 = Wg[(lm-8)*4 + k0 + 1]; }
    v2f aw; aw[0] = a0; aw[1] = a1;   // A[m=lm][k0..k0+1], 16x4 stacked weights

    for (int cc = 0; cc < NKP/16; ++cc) {
        const int kid = cc*16 + lm;
        const int kr = kid / KRR, kc = kid % KRR;
        const int gh = h0 - PADR + kr, gw = w0 - PADR + kc;
        float b0 = 0.f, b1 = 0.f;
        if (kid < NKEY && gh >= 0 && gh < HH && gw >= 0 && gw < WW) {
            const float* px = x + (((size_t)bb*NC + k0) * HH + gh) * WW + gw;
            b0 = px[0];
            b1 = px[(size_t)HH * WW];     // next channel plane
        }
        v2f bx; bx[0] = b0; bx[1] = b1;   // B[k0..k0+1][n=lm] = x[c][key]
        v8f d = __builtin_amdgcn_wmma_f32_16x16x4_f32(false, aw, false, bx,
                                                      (short)0, cz, false, false);
        // D[m = v + 8*hi][n = lm]; rows 0-3 theta, 4-7 phi, 8-11 g
        if (hi == 0) {
            th[0][kid] = d[0]; th[1][kid] = d[1]; th[2][kid] = d[2]; th[3][kid] = d[3];
            ph[0][kid] = d[4]; ph[1][kid] = d[5]; ph[2][kid] = d[6]; ph[3][kid] = d[7];
        } else {
            gg[0][kid] = d[0]; gg[1][kid] = d[1]; gg[2][kid] = d[2]; gg[3][kid] = d[3];
        }
    }
    __syncthreads();

    // ---- Stage B: S[q][k] = theta[:,q] . phi[:,k], masked; track row max ----
    float rm[8];
#pragma unroll
    for (int v = 0; v < 8; ++v) rm[v] = -3.0e38f;

    const int qkidx = ((lm >> 2) + PADR) * KRR + (lm & 3) + PADR;  // query's key idx
    v2f at; at[0] = th[k0][qkidx]; at[1] = th[k0+1][qkidx];        // A[m=q][c]
    for (int cc = 0; cc < NKP/16; ++cc) {
        const int kid = cc*16 + lm;
        v2f bp; bp[0] = ph[k0][kid]; bp[1] = ph[k0+1][kid];        // B[c][n=key]
        v8f d = __builtin_amdgcn_wmma_f32_16x16x4_f32(false, at, false, bp,
                                                      (short)0, cz, false, false);
        const int kr = kid / KRR, kc = kid % KRR;
        const bool kv = kid < NKEY;
#pragma unroll
        for (int v = 0; v < 8; ++v) {
            const int q  = v + 8*hi;
            const int dr = kr - ((q >> 2) + PADR);
            const int dc = kc - ((q &  3) + PADR);
            const bool ok = kv && dr >= -PADR && dr <= PADR
                               && dc >= -PADR && dc <= PADR;
            const float val = ok ? d[v] : -1.0e30f;
            sc[q][kid] = val;
            rm[v] = fmaxf(rm[v], val);
        }
    }

    // Row max: reduce across the 16 lanes of this half (rows m = v + 8*hi).
#pragma unroll
    for (int m = 1; m < 16; m <<= 1) {
#pragma unroll
        for (int v = 0; v < 8; ++v) rm[v] = fmaxf(rm[v], __shfl_xor(rm[v], m, 32));
    }

    // ---- Stage C: exponentiate in place (all 32 lanes; each touches only the
    //      columns it wrote, so intra-wave LDS ordering suffices), track row sums.
    float rs[8];
#pragma unroll
    for (int v = 0; v < 8; ++v) rs[v] = 0.f;

    for (int cc = 0; cc < NKP/16; ++cc) {
        const int col = cc*16 + lm;
#pragma unroll
        for (int v = 0; v < 8; ++v) {
            const int q = v + 8*hi;
            const float e = __expf(sc[q][col] - rm[v]);
            sc[q][col] = e;              // unnormalized attention weight
            rs[v] += e;
        }
    }
#pragma unroll
    for (int m = 1; m < 16; m <<= 1) {
#pragma unroll
        for (int v = 0; v < 8; ++v) rs[v] += __shfl_xor(rs[v], m, 32);
    }
    float inv[8];
#pragma unroll
    for (int v = 0; v < 8; ++v) inv[v] = 1.0f / rs[v];
    __syncthreads();

    // ---- Stage D: O^T[q][c] = sum_k e[q][k] * g[c][k]  (accumulating WMMAs),
    //      normalization folded into the accumulator scale afterwards. ----
    v8f acc = {};
    const float cf = (lm < 4) ? 1.0f : 0.0f;
    const int   gc = lm & 3;
    for (int t4 = 0; t4 < NKP/4; ++t4) {
        const int kb = t4*4 + k0;
        v2f aa; aa[0] = sc[lm][kb];        aa[1] = sc[lm][kb+1];      // A[q][key]
        v2f bg; bg[0] = cf * gg[gc][kb];   bg[1] = cf * gg[gc][kb+1]; // B[key][c]
        acc = __builtin_amdgcn_wmma_f32_16x16x4_f32(false, aa, false, bg,
                                                    (short)0, acc, false, false);
    }
    if (lm < 4) {   // D[m=q][n=c]: lane owns channel lm, 8 queries in acc
#pragma unroll
        for (int v = 0; v < 8; ++v) ov[lm][v + 8*hi] = acc[v] * inv[v];
    }
    __syncthreads();

    // ---- Stage E: result = Ww @ O + x ----
    float wa0 = 0.f, wa1 = 0.f;
    if (lm < 4) { wa0 = Ww[lm*4 + k0]; wa1 = Ww[lm*4 + k0 + 1]; }
    v2f awf; awf[0] = wa0; awf[1] = wa1;                 // A[o][c] (rows 0-3)
    v2f bo;  bo[0] = ov[k0][lm]; bo[1] = ov[k0+1][lm];   // B[c][q]
    v8f r = __builtin_amdgcn_wmma_f32_16x16x4_f32(false, awf, false, bo,
                                                  (short)0, cz, false, false);
    if (hi == 0) {  // rows 0-3 live in v=0..3 of the low half
        const int q  = lm;
        const int gh = h0 + (q >> 2), gw = w0 + (q & 3);
#pragma unroll
        for (int o = 0; o < 4; ++o) {
            const size_t idx = (((size_t)bb*NC + o) * HH + gh) * WW + gw;
            out[idx] = r[o] + x[idx];
        }
    }
}

extern "C" void kernel_launch(void* const* d_in, const int* in_sizes, int n_in,
                              void* d_out, int out_size, void* d_ws, size_t ws_size,
                              hipStream_t stream) {
    const float* x  = (const float*)d_in[0];
    const float* Wt = (const float*)d_in[1];
    const float* Wp = (const float*)d_in[2];
    const float* Wg = (const float*)d_in[3];
    const float* Ww = (const float*)d_in[4];
    float* out = (float*)d_out;

    dim3 grid(NB * (HH/QT) * (WW/QT));   // 4 * 64 * 64 = 16384 tiles
    dim3 block(32);                      // one wave32 per tile
    hipLaunchKernelGGL(blocknl_wmma_kernel, grid, block, 0, stream,
                       x, Wt, Wp, Wg, Ww, out);
}